// HierarchicalSoftmax_10290741641971
// MI455X (gfx1250) — compile-verified
//
#include <hip/hip_runtime.h>
#include <math.h>

// CDNA5 (gfx1250) wave32 WMMA types
typedef __attribute__((ext_vector_type(2))) float v2f;   // A/B operand of V_WMMA_F32_16X16X4_F32
typedef __attribute__((ext_vector_type(4))) float v4f;   // 16B vector load (native clang vector)
typedef __attribute__((ext_vector_type(8))) float v8f;   // 16x16 f32 C/D accumulator

__device__ __forceinline__ float log_sigmoid_f(float z) {
    // numerically stable: min(z,0) - log1p(exp(-|z|))
    return fminf(z, 0.0f) - log1pf(expf(-fabsf(z)));
}

__global__ __launch_bounds__(256)
void HierSoftmax_wmma_f32_kernel(const float* __restrict__ x,      // [B, D]
                                 const float* __restrict__ W,      // [V-1, D]
                                 const int*   __restrict__ nodes,  // [V, Dmax]
                                 const float* __restrict__ signs,  // [V, Dmax]
                                 const int*   __restrict__ target, // [B]
                                 float* __restrict__ out,          // [B]
                                 int B, int D, int Dmax)
{
    const int wave = blockIdx.x * (blockDim.x >> 5) + (threadIdx.x >> 5);
    if (wave >= B) return;                       // wave-uniform: EXEC stays all-1 for WMMA

    const int lane = threadIdx.x & 31;
    const int h    = lane >> 4;                  // half-wave: selects K sub-group (+4 elements)
    const int r    = lane & 15;                  // A-matrix row (Huffman depth slot)

    const int t = target[wave];
    const float* __restrict__ xb = x + (size_t)wave * D;

    // Row gather: lane r streams inner_weights[nodes[t][r]] (pad slots -> row 0, masked later)
    int node_r = 0;
    if (r < Dmax) node_r = nodes[(size_t)t * Dmax + r];
    const float* __restrict__ wrow = W + (size_t)node_r * D;

    v8f acc = {0.f, 0.f, 0.f, 0.f, 0.f, 0.f, 0.f, 0.f};

    const int koff = 4 * h;                      // lane-half K offset inside each K=8 chunk

    // K loop: 2x V_WMMA_F32_16X16X4_F32 per 8-wide chunk. Unroll 4 so the scheduler can
    // cluster 8 global_load_b128 ahead of 8 WMMAs (memory-level parallelism per wave).
    #pragma unroll 4
    for (int k0 = 0; k0 < D; k0 += 8) {
        // W rows: regular temporal (heavy cross-wave reuse at shallow Huffman depths;
        // whole 206MB table nearly fits the 192MB L2).
        v4f aq = *(const v4f*)(wrow + k0 + koff);
        // x[b]: touched by exactly one wave -> non-temporal, don't pollute L2.
        v4f bq = __builtin_nontemporal_load((const v4f*)(xb + k0 + koff));

        v2f a0; a0[0] = aq[0]; a0[1] = aq[1];
        v2f b0; b0[0] = bq[0]; b0[1] = bq[1];
        acc = __builtin_amdgcn_wmma_f32_16x16x4_f32(false, a0, false, b0, (short)0, acc, false, false);
        v2f a1; a1[0] = aq[2]; a1[1] = aq[3];
        v2f b1; b1[0] = bq[2]; b1[1] = bq[3];
        acc = __builtin_amdgcn_wmma_f32_16x16x4_f32(false, a1, false, b1, (short)0, acc, false, false);
    }

    // D columns are identical (B columns were x-broadcast). Lane-half h holds scores for
    // depths m = 8*h + v in acc[v] (C/D layout: VGPR v, lanes 0-15 -> M=v, lanes 16-31 -> M=v+8).
    float partial = 0.f;
    #pragma unroll
    for (int v = 0; v < 8; ++v) {
        const int m = 8 * h + v;
        float s = 0.f;
        if (m < Dmax) s = signs[(size_t)t * Dmax + m];
        if (s != 0.f) partial += log_sigmoid_f(s * acc[v]);
    }
    float total = partial + __shfl_xor(partial, 16, 32);

    // Generic tail for Dmax > 16 (does not fire for uniform Huffman over 50257 leaves)
    for (int m = 16; m < Dmax; ++m) {
        const float s = signs[(size_t)t * Dmax + m];
        if (s != 0.f) {
            const int nd = nodes[(size_t)t * Dmax + m];
            const float* __restrict__ wr = W + (size_t)nd * D;
            float d = 0.f;
            for (int j = lane; j < D; j += 32) d += wr[j] * xb[j];
            d += __shfl_xor(d, 16, 32);
            d += __shfl_xor(d, 8, 32);
            d += __shfl_xor(d, 4, 32);
            d += __shfl_xor(d, 2, 32);
            d += __shfl_xor(d, 1, 32);
            total += log_sigmoid_f(s * d);
        }
    }

    if (lane == 0) __builtin_nontemporal_store(total, out + wave);
}

extern "C" void kernel_launch(void* const* d_in, const int* in_sizes, int n_in,
                              void* d_out, int out_size, void* d_ws, size_t ws_size,
                              hipStream_t stream) {
    const float* x      = (const float*)d_in[0];   // [B, D] fp32
    const float* W      = (const float*)d_in[1];   // [V-1, D] fp32
    const int*   nodes  = (const int*)  d_in[2];   // [V, Dmax] int32
    const float* signs  = (const float*)d_in[3];   // [V, Dmax] fp32
    const int*   target = (const int*)  d_in[4];   // [B] int32
    float*       out    = (float*)d_out;           // [B] fp32

    const int B       = in_sizes[4];
    const int D       = in_sizes[0] / B;           // 1024 (multiple of 8 assumed)
    const int n_inner = in_sizes[1] / D;
    const int V       = n_inner + 1;
    const int Dmax    = in_sizes[2] / V;           // 16 for uniform Huffman over 50257

    const int WAVES_PER_BLOCK = 8;                 // 256 threads = 8 wave32
    const int threads = WAVES_PER_BLOCK * 32;
    const int blocks  = (B + WAVES_PER_BLOCK - 1) / WAVES_PER_BLOCK;
    HierSoftmax_wmma_f32_kernel<<<blocks, threads, 0, stream>>>(
        x, W, nodes, signs, target, out, B, D, Dmax);
}